// ComposePGT_9191230013554
// MI455X (gfx1250) — compile-verified
//
#include <hip/hip_runtime.h>

// ---------------------------------------------------------------------------
// ComposePGT for MI455X (gfx1250, wave32).
// Memory-bound workload (~0.5 FLOP/byte). Strategy: few fused coalesced
// passes, u8 intermediates, LDS histograms, and exact f32 WMMA
// (V_WMMA_F32_16X16X4_F32) for the 256-bin cumulative sums (integer counts
// < 2^24 => f32 matmul accumulation is bit-exact).
// ---------------------------------------------------------------------------

#define BATCH 16
#define HH 512
#define WW 512
#define HWPIX (HH * WW)              // 262144
#define NBINS 256
#define MARGIN 12                    // eye expand margin
#define NMATCH 4                     // skin(m1), m4, lip(m0), eye(m_se)

typedef unsigned char u8;
typedef unsigned int u32;

typedef __attribute__((ext_vector_type(2))) float v2f;
typedef __attribute__((ext_vector_type(8))) float v8f;

#if __has_builtin(__builtin_amdgcn_wmma_f32_16x16x4_f32)
#define HAVE_WMMA_F32X4 1
#endif

__device__ __forceinline__ float clamp01(float x) {
    return fminf(fmaxf(x, 0.0f), 1.0f);
}

// ---------------------------------------------------------------------------
// 0) zero the histogram region of the workspace
// ---------------------------------------------------------------------------
__global__ void zero_u32_kernel(u32* __restrict__ p, int n) {
    int i = blockIdx.x * blockDim.x + threadIdx.x;
    if (i < n) p[i] = 0u;
}

// ---------------------------------------------------------------------------
// 1) horizontal max-pool (k = 25, SAME) of mask[2]+mask[3].
//    One block per image row; values {0,1,2} stored as u8.
// ---------------------------------------------------------------------------
__global__ void expand_h_kernel(const float* __restrict__ masks /*(B,5,H,W)*/,
                                u8* __restrict__ tmp /*(B,H,W)*/) {
    const int row = blockIdx.x;              // 0 .. B*H-1
    const int b = row / HH;
    const int y = row % HH;
    const float* m2 = masks + ((size_t)b * 5 + 2) * HWPIX + (size_t)y * WW;
    const float* m3 = masks + ((size_t)b * 5 + 3) * HWPIX + (size_t)y * WW;

    __shared__ u8 vals[WW];
    for (int x = threadIdx.x; x < WW; x += blockDim.x)
        vals[x] = (u8)(m2[x] + m3[x]);       // exact: {0,1,2}
    __syncthreads();

    u8* o = tmp + (size_t)row * WW;
    for (int x = threadIdx.x; x < WW; x += blockDim.x) {
        int lo = x - MARGIN; if (lo < 0) lo = 0;
        int hi = x + MARGIN; if (hi > WW - 1) hi = WW - 1;
        u8 mx = 0;
        for (int j = lo; j <= hi; ++j) { u8 v = vals[j]; mx = (v > mx) ? v : mx; }
        o[x] = mx;
    }
}

// ---------------------------------------------------------------------------
// 2) vertical max-pool (k = 25, SAME) of tmp, multiplied by mask[1].
// ---------------------------------------------------------------------------
__global__ void expand_v_kernel(const u8* __restrict__ tmp /*(B,H,W)*/,
                                const float* __restrict__ masks /*(B,5,H,W)*/,
                                u8* __restrict__ outp /*(B,H,W)*/) {
    const int b = blockIdx.y;
    const int p = blockIdx.x * blockDim.x + threadIdx.x;   // 0 .. HW-1
    const int y = p / WW;
    const int x = p - y * WW;
    const u8* t = tmp + (size_t)b * HWPIX;
    int lo = y - MARGIN; if (lo < 0) lo = 0;
    int hi = y + MARGIN; if (hi > HH - 1) hi = HH - 1;
    u8 mx = 0;
    for (int j = lo; j <= hi; ++j) { u8 v = t[j * WW + x]; mx = (v > mx) ? v : mx; }
    float m1 = masks[((size_t)b * 5 + 1) * HWPIX + p];
    outp[(size_t)b * HWPIX + p] = (m1 > 0.0f) ? mx : (u8)0;
}

// ---------------------------------------------------------------------------
// 3) 5x5 box-count (SAME, zero pad) of m_se -> integer counts 0..50 in u8.
// ---------------------------------------------------------------------------
__global__ void blur5_kernel(const u8* __restrict__ mse /*(B,H,W)*/,
                             u8* __restrict__ mbc /*(B,H,W)*/) {
    const int b = blockIdx.y;
    const int p = blockIdx.x * blockDim.x + threadIdx.x;
    const int y = p / WW;
    const int x = p - y * WW;
    const u8* s = mse + (size_t)b * HWPIX;
    int sum = 0;
    for (int dy = -2; dy <= 2; ++dy) {
        int yy = y + dy;
        if (yy < 0 || yy >= HH) continue;
        for (int dx = -2; dx <= 2; ++dx) {
            int xx = x + dx;
            if (xx < 0 || xx >= WW) continue;
            sum += s[yy * WW + xx];
        }
    }
    mbc[(size_t)b * HWPIX + p] = (u8)sum;
}

// ---------------------------------------------------------------------------
// 4) weighted histograms. combo = b*8 + match*2 + side; LDS u32[3][256].
//    Weight = mask value (1 or 2); bin from de_norm(img)*255*mask.
// ---------------------------------------------------------------------------
__global__ void hist_kernel(const float* __restrict__ srcs,
                            const float* __restrict__ tars,
                            const float* __restrict__ msrc,
                            const float* __restrict__ mtar,
                            const u8* __restrict__ mse,
                            const u8* __restrict__ mre,
                            u32* __restrict__ hist) {
    __shared__ u32 lh[3 * NBINS];
    for (int i = threadIdx.x; i < 3 * NBINS; i += blockDim.x) lh[i] = 0u;
    __syncthreads();

    const int combo = blockIdx.y;            // 0 .. B*8-1
    const int side  = combo & 1;             // 0 = source, 1 = reference
    const int match = (combo >> 1) & 3;
    const int b     = combo >> 3;

    const float* img = (side ? tars : srcs) + (size_t)b * 3 * HWPIX;
    const float* mp = nullptr;
    const u8* ep = nullptr;
    if (match < 3) {
        int mi = (match == 0) ? 1 : ((match == 1) ? 4 : 0);
        mp = (side ? mtar : msrc) + ((size_t)b * 5 + mi) * HWPIX;
    } else {
        ep = (side ? mre : mse) + (size_t)b * HWPIX;
    }

    const int chunk = HWPIX / gridDim.x;
    const int start = blockIdx.x * chunk;
    const int end = start + chunk;
    for (int p = start + threadIdx.x; p < end; p += blockDim.x) {
        float m = (match < 3) ? mp[p] : (float)ep[p];
        if (m > 0.0f) {
            u32 w = (u32)m;                  // 1 or 2, exact
#pragma unroll
            for (int c = 0; c < 3; ++c) {
                float x = img[c * HWPIX + p];
                float d = clamp01((x + 1.0f) * 0.5f) * 255.0f * m;
                int bin = (int)d; if (bin > NBINS - 1) bin = NBINS - 1;
                atomicAdd(&lh[c * NBINS + bin], w);
            }
        }
    }
    __syncthreads();

    u32* gh = hist + (((size_t)(b * NMATCH + match) * 2 + side) * 3) * NBINS;
    for (int i = threadIdx.x; i < 3 * NBINS; i += blockDim.x) {
        u32 v = lh[i];
        if (v) atomicAdd(&gh[i], v);
    }
}

// ---------------------------------------------------------------------------
// 5) CDF + searchsorted table. One wave (32 lanes) per (b, match, channel).
//    cumsum(256) done as H(16x16) x upper-triangular-ones via chained
//    V_WMMA_F32_16X16X4_F32 (exact for integer counts < 2^24).
// ---------------------------------------------------------------------------
__device__ __forceinline__ void wave_rowprefix256(const float* __restrict__ h,
                                                  float* __restrict__ p,
                                                  float* __restrict__ off,
                                                  int lane) {
#if defined(HAVE_WMMA_F32X4)
    const int hi = (lane >= 16) ? 1 : 0;
    const int m = lane & 15;
    v8f acc = {0.f, 0.f, 0.f, 0.f, 0.f, 0.f, 0.f, 0.f};
#pragma unroll
    for (int k0 = 0; k0 < 16; k0 += 4) {
        const int kA = k0 + (hi ? 2 : 0);
        v2f a, bb;
        a.x = h[m * 16 + kA];
        a.y = h[m * 16 + kA + 1];
        bb.x = (kA <= m) ? 1.0f : 0.0f;      // T[k][n] = (k <= n)
        bb.y = (kA + 1 <= m) ? 1.0f : 0.0f;
        acc = __builtin_amdgcn_wmma_f32_16x16x4_f32(
            /*neg_a=*/false, a, /*neg_b=*/false, bb,
            /*c_mod=*/(short)0, acc, /*reuse_a=*/false, /*reuse_b=*/false);
    }
#pragma unroll
    for (int v = 0; v < 8; ++v)              // D layout: VGPR v -> rows v / v+8
        p[(v + hi * 8) * 16 + m] = acc[v];
#else
    // scalar fallback (still exact)
    if (lane < 16) {
        float run = 0.0f;
        for (int k = 0; k < 16; ++k) { run += h[lane * 16 + k]; p[lane * 16 + k] = run; }
    }
#endif
    __syncthreads();
    if (lane == 0) {
        float run = 0.0f;
        for (int r = 0; r < 16; ++r) { off[r] = run; run += p[r * 16 + 15]; }
        off[16] = run;                        // grand total = sum(h), exact
    }
    __syncthreads();
}

__global__ void __launch_bounds__(32)
cdf_table_kernel(const u32* __restrict__ hist, float* __restrict__ tables) {
    const int id = blockIdx.x;                // 0 .. B*4*3-1
    const int b = id / (NMATCH * 3);
    const int rem = id - b * (NMATCH * 3);
    const int match = rem / 3;
    const int c = rem - match * 3;
    const int lane = threadIdx.x;

    __shared__ float hA[NBINS];
    __shared__ float pA[NBINS];
    __shared__ float cdfS[NBINS];
    __shared__ float cdfR[NBINS];
    __shared__ float offA[17];

    const u32* ghS = hist + (((size_t)(b * NMATCH + match) * 2 + 0) * 3 + c) * NBINS;
    const u32* ghR = hist + (((size_t)(b * NMATCH + match) * 2 + 1) * 3 + c) * NBINS;

    // ---- source CDF ----
    for (int k = 0; k < 8; ++k) { int i = lane + 32 * k; hA[i] = (float)ghS[i]; }
    __syncthreads();
    wave_rowprefix256(hA, pA, offA, lane);
    {
        float denom = fmaxf(offA[16], 1e-6f);
        for (int k = 0; k < 8; ++k) {
            int i = lane + 32 * k;
            cdfS[i] = (pA[i] + offA[i >> 4]) / denom;
        }
    }
    __syncthreads();

    // ---- reference CDF ----
    for (int k = 0; k < 8; ++k) { int i = lane + 32 * k; hA[i] = (float)ghR[i]; }
    __syncthreads();
    wave_rowprefix256(hA, pA, offA, lane);
    {
        float denom = fmaxf(offA[16], 1e-6f);
        for (int k = 0; k < 8; ++k) {
            int i = lane + 32 * k;
            cdfR[i] = (pA[i] + offA[i >> 4]) / denom;
        }
    }
    __syncthreads();

    // ---- table[i] = clip(searchsorted(cdf_r, cdf_s[i]), 0, 255) ----
    float* tout = tables + (((size_t)b * NMATCH + match) * 3 + c) * NBINS;
    for (int k = 0; k < 8; ++k) {
        int i = lane + 32 * k;
        float v = cdfS[i];
        int cnt = 0;
        for (int j = 0; j < NBINS; ++j) cnt += (cdfR[j] < v) ? 1 : 0;
        if (cnt > NBINS - 1) cnt = NBINS - 1;
        tout[i] = (float)cnt;
    }
}

// ---------------------------------------------------------------------------
// 6) fused compose: all matches + eye blur blend + three alpha blends.
//    12 tables (12 KB) staged in LDS per batch item.
// ---------------------------------------------------------------------------
__global__ void compose_kernel(const float* __restrict__ srcs,
                               const float* __restrict__ tars,
                               const float* __restrict__ msrc,
                               const u8* __restrict__ mse,
                               const u8* __restrict__ mbc,
                               const float* __restrict__ tables,
                               float* __restrict__ out) {
    __shared__ float tab[12 * NBINS];
    const int b = blockIdx.y;
    const float* tb = tables + (size_t)b * 12 * NBINS;
    for (int i = threadIdx.x; i < 12 * NBINS; i += blockDim.x) tab[i] = tb[i];
    __syncthreads();

    const float* is = srcs + (size_t)b * 3 * HWPIX;
    const float* ir = tars + (size_t)b * 3 * HWPIX;
    const float* m1p = msrc + ((size_t)b * 5 + 1) * HWPIX;
    const float* m4p = msrc + ((size_t)b * 5 + 4) * HWPIX;
    const float* m0p = msrc + ((size_t)b * 5 + 0) * HWPIX;
    const u8* msep = mse + (size_t)b * HWPIX;
    const u8* mbp = mbc + (size_t)b * HWPIX;
    float* op = out + (size_t)b * 3 * HWPIX;

    const int chunk = HWPIX / gridDim.x;
    const int start = blockIdx.x * chunk;
    const int end = start + chunk;
    for (int p = start + threadIdx.x; p < end; p += blockDim.x) {
        // speculative prefetch of the next strip (gfx1250 global_prefetch_b8)
        __builtin_prefetch(&is[p + (int)blockDim.x], 0, 1);
        __builtin_prefetch(&ir[p + (int)blockDim.x], 0, 1);

        float m1 = m1p[p];
        float m4 = m4p[p];
        float m0 = m0p[p];
        float mseV = (float)msep[p];                       // 0, 1 or 2
        float mb = ((float)mbp[p] / 25.0f) * mseV;         // mask_blur value

#pragma unroll
        for (int c = 0; c < 3; ++c) {
            float x = is[c * HWPIX + p];
            float rr = ir[c * HWPIX + p];
            float d = clamp01((x + 1.0f) * 0.5f) * 255.0f;
            int bn = (int)d; if (bn > NBINS - 1) bn = NBINS - 1;

            float pgt = x;
            if (m1 > 0.0f) pgt = tab[(0 * 3 + c) * NBINS + bn] / 255.0f * 2.0f - 1.0f;
            if (m4 > 0.0f) pgt = tab[(1 * 3 + c) * NBINS + bn] / 255.0f * 2.0f - 1.0f;
            if (m0 > 0.0f) pgt = tab[(2 * 3 + c) * NBINS + bn] / 255.0f * 2.0f - 1.0f;

            // eye: bin uses d * m_se (m_se may be 2); blend with fractional mb
            float se = d * mseV;
            int be = (int)se; if (be > NBINS - 1) be = NBINS - 1;
            float eyeVal = (mseV > 0.0f) ? tab[(3 * 3 + c) * NBINS + be] : se;
            float eye = eyeVal / 255.0f * 2.0f - 1.0f;
            pgt = (1.0f - mb) * pgt + mb * eye;

            // final alpha blends toward reference image
            pgt = (1.0f - 0.3f * m1) * pgt + 0.3f * m1 * rr;
            pgt = (1.0f - 0.8f * mseV) * pgt + 0.8f * mseV * rr;
            pgt = (1.0f - 0.1f * m0) * pgt + 0.1f * m0 * rr;

            op[c * HWPIX + p] = pgt;
        }
    }
}

// ---------------------------------------------------------------------------
// launch
// ---------------------------------------------------------------------------
extern "C" void kernel_launch(void* const* d_in, const int* in_sizes, int n_in,
                              void* d_out, int out_size, void* d_ws, size_t ws_size,
                              hipStream_t stream) {
    (void)in_sizes; (void)n_in; (void)out_size; (void)ws_size;
    const float* srcs = (const float*)d_in[0];   // (B,3,H,W)
    const float* tars = (const float*)d_in[1];   // (B,3,H,W)
    const float* msrc = (const float*)d_in[2];   // (B,5,H,W)
    const float* mtar = (const float*)d_in[3];   // (B,5,H,W)
    // d_in[4], d_in[5] (landmarks) are unused by the reference math.
    float* out = (float*)d_out;

    char* ws = (char*)d_ws;
    const size_t HWB = (size_t)BATCH * HWPIX;    // 4,194,304
    u8* tmp = (u8*)ws;                           // (B,H,W) u8
    u8* mse = (u8*)(ws + HWB);                   // (B,H,W) u8
    u8* mre = (u8*)(ws + 2 * HWB);               // (B,H,W) u8
    u8* mbc = (u8*)(ws + 3 * HWB);               // (B,H,W) u8
    u32* hist = (u32*)(ws + 4 * HWB);            // 384*256 u32
    float* tables = (float*)(ws + 4 * HWB + (size_t)BATCH * NMATCH * 2 * 3 * NBINS * 4);

    const int nhist = BATCH * NMATCH * 2 * 3 * NBINS;   // 98304
    zero_u32_kernel<<<(nhist + 255) / 256, 256, 0, stream>>>(hist, nhist);

    // eye-region expansion (25x25 max pool, separable) for source and ref
    expand_h_kernel<<<BATCH * HH, 256, 0, stream>>>(msrc, tmp);
    expand_v_kernel<<<dim3(HWPIX / 256, BATCH), 256, 0, stream>>>(tmp, msrc, mse);
    expand_h_kernel<<<BATCH * HH, 256, 0, stream>>>(mtar, tmp);
    expand_v_kernel<<<dim3(HWPIX / 256, BATCH), 256, 0, stream>>>(tmp, mtar, mre);

    // 5x5 blur counts for the eye blend
    blur5_kernel<<<dim3(HWPIX / 256, BATCH), 256, 0, stream>>>(mse, mbc);

    // weighted histograms for all (b, match, side)
    hist_kernel<<<dim3(32, BATCH * NMATCH * 2), 256, 0, stream>>>(
        srcs, tars, msrc, mtar, mse, mre, hist);

    // CDFs (WMMA cumsum) + match tables
    cdf_table_kernel<<<BATCH * NMATCH * 3, 32, 0, stream>>>(hist, tables);

    // fused composition
    compose_kernel<<<dim3(64, BATCH), 256, 0, stream>>>(
        srcs, tars, msrc, mse, mbc, tables, out);
}